// TemporalFeatureGraph_43413529428135
// MI455X (gfx1250) — compile-verified
//
#include <hip/hip_runtime.h>

typedef __attribute__((ext_vector_type(2))) float v2f;
typedef __attribute__((ext_vector_type(8))) float v8f;
typedef __attribute__((ext_vector_type(4))) unsigned int tdm_g0_t;
typedef __attribute__((ext_vector_type(8))) int tdm_g1_t;
typedef __attribute__((ext_vector_type(4))) int tdm_g2_t;
typedef __attribute__((ext_vector_type(8))) int tdm_g4_t;

#define BATCH 8
#define VLEN  16
#define CH    64
#define HW    784
#define NN    (VLEN * HW)      /* 12544 nodes per sample */
#define T1    (VLEN - 1)       /* 15 frame pairs */
#define KTOP  4
#define NEDGE (2 * T1 * KTOP)  /* 120 directed edges per sample */
#define BN_EPS 1e-5f

// ---------------------------------------------------------------------------
// TDM: 1-D contiguous tile load, global -> LDS. D# per CDNA5 ISA:
//  group0: count=1 | lds_addr | global_addr[56:0] | type=2
//  group1: data_size=2 (4B), tensor_dim0 = tile_dim0 = nelem, strides linear
//  groups 2/3/4: zero (tile_dim2/3/4 = 0 -> dims unused, pure 1-D)
// 6-arg clang-23 builtin form. Issued by wave 0 only (TENSOR ops ignore EXEC
// but are per-wave), waited with s_wait_tensorcnt, then published via barrier.
// ---------------------------------------------------------------------------
__device__ __forceinline__ void tdm_load_1d(unsigned lds_off, const void* gptr,
                                            unsigned nelem)
{
    const unsigned long long ga = (unsigned long long)gptr;
    tdm_g0_t g0;
    g0.x = 1u;                                        // count=1, user mode
    g0.y = lds_off;                                   // lds_addr (bytes)
    g0.z = (unsigned)(ga & 0xFFFFFFFFu);              // global_addr[31:0]
    g0.w = (unsigned)((ga >> 32) & 0x01FFFFFFu) |     // global_addr[56:32]
           (2u << 30);                                // type=2 ("image")
    tdm_g1_t g1;
    g1[0] = (int)(2u << 16);                          // wg_mask=0, data_size=4B
    g1[1] = (int)((nelem & 0xFFFFu) << 16);           // tensor_dim0[15:0]
    g1[2] = (int)(nelem >> 16);                       // tensor_dim0[31:16], dim1=0
    g1[3] = (int)((nelem & 0xFFFFu) << 16);           // dim1 hi=0, tile_dim0=nelem
    g1[4] = 0;                                        // tile_dim1=0, tile_dim2=0
    g1[5] = (int)nelem;                               // tensor_dim0_stride[31:0]
    g1[6] = 0;                                        // stride0 hi, stride1 lo
    g1[7] = 0;                                        // stride1 hi
    tdm_g2_t gz4 = {0, 0, 0, 0};
    tdm_g4_t gz8 = {0, 0, 0, 0, 0, 0, 0, 0};
    __builtin_amdgcn_tensor_load_to_lds(g0, g1, gz4, gz4, gz8, 0);
}

// ---------------------------------------------------------------------------
// K1: temporal conv3d(3x1x1, pad 1) + BN(eval).  x[tlen,C,HW] -> xd[b,v,hw,c]
// 256 threads / 8 waves per block; block covers 7 M-tiles x 4 N-tiles of one
// (b, v); weights staged in LDS by TDM, B-fragments via ds_load.
// ---------------------------------------------------------------------------
__global__ __launch_bounds__(256) void conv_bn_down_kernel(
    const float* __restrict__ x, const float* __restrict__ w,
    const float* __restrict__ gamma, const float* __restrict__ beta,
    const float* __restrict__ mean, const float* __restrict__ var,
    float* __restrict__ xd)
{
    __shared__ float lw[CH * CH * 3];                 // 48 KB of 320 KB/WGP
    if (threadIdx.x < 32) {
        tdm_load_1d((unsigned)(size_t)lw, w, CH * CH * 3);
        __builtin_amdgcn_s_wait_tensorcnt(0);
    }
    __syncthreads();

    const int lane = threadIdx.x & 31;
    const int wave = threadIdx.x >> 5;
    int blk = blockIdx.x;                             // (b, v, mgroup)
    const int mg = blk % 7;  blk /= 7;
    const int v  = blk % VLEN; blk /= VLEN;
    const int b  = blk;
    const int m    = lane & 15;                       // A row
    const int koff = (lane >> 4) << 1;                // K pair select
    const int col  = lane & 15;                       // B/D column

    for (int tile = wave; tile < 7 * 4; tile += 8) {
        const int mt = mg * 7 + (tile >> 2);
        const int n0 = (tile & 3) * 16;
        const int p0 = mt * 16;
        v8f acc = {};
        for (int dt = 0; dt < 3; ++dt) {
            const int vv = v + dt - 1;                // zero pad temporal edges
            if (vv < 0 || vv >= VLEN) continue;
            const float* xp = x + (size_t)(b * VLEN + vv) * CH * HW + (p0 + m);
            const float* wp = lw + (n0 + col) * CH * 3 + dt;
            for (int k = 0; k < CH; k += 4) {
                v2f a, bm;
                a.x  = xp[(size_t)(k + koff) * HW];
                a.y  = xp[(size_t)(k + koff + 1) * HW];
                bm.x = wp[(k + koff) * 3];            // ds_load from staged w
                bm.y = wp[(k + koff + 1) * 3];
                acc = __builtin_amdgcn_wmma_f32_16x16x4_f32(
                    false, a, false, bm, (short)0, acc, false, false);
            }
        }
        const int c = n0 + col;
        const float s = gamma[c] * rsqrtf(var[c] + BN_EPS);
        const float t = beta[c] - mean[c] * s;
        float* op = xd + (((size_t)(b * VLEN + v)) * HW + p0) * CH + c;
#pragma unroll
        for (int r = 0; r < 8; ++r) {
            const int row = r + ((lane >> 4) << 3);   // D: VGPR r -> rows r/r+8
            op[(size_t)row * CH] = acc[r] * s + t;
        }
    }
}

// ---------------------------------------------------------------------------
// K2: xw = nodes @ gcn_w^T.  xd[b,n,c] -> xw[b,n,c].  gcn_w staged via TDM.
// Block: 256 threads, 4 M-tiles x 4 N-tiles.
// ---------------------------------------------------------------------------
__global__ __launch_bounds__(256) void gemm_xw_kernel(
    const float* __restrict__ xd, const float* __restrict__ gw,
    float* __restrict__ xw)
{
    __shared__ float lg[CH * CH];                     // 16 KB
    if (threadIdx.x < 32) {
        tdm_load_1d((unsigned)(size_t)lg, gw, CH * CH);
        __builtin_amdgcn_s_wait_tensorcnt(0);
    }
    __syncthreads();

    const int lane = threadIdx.x & 31;
    const int wave = threadIdx.x >> 5;
    const int mg = blockIdx.x % (NN / 16 / 4);        // 196 groups of 4 M-tiles
    const int b  = blockIdx.x / (NN / 16 / 4);
    const int m    = lane & 15;
    const int koff = (lane >> 4) << 1;
    const int col  = lane & 15;

    for (int tile = wave; tile < 4 * 4; tile += 8) {
        const int mt = mg * 4 + (tile >> 2);
        const int n0 = (tile & 3) * 16;
        const float* ap = xd + ((size_t)b * NN + (size_t)mt * 16 + m) * CH;
        const float* bp = lg + (n0 + col) * CH;       // B[k][n] = gw[n*CH+k]
        __builtin_prefetch(ap + 16 * CH, 0, 1);       // global_prefetch_b8

        v8f acc = {};
#pragma unroll
        for (int k = 0; k < CH; k += 4) {
            v2f a  = *(const v2f*)(ap + k + koff);    // 8B-aligned global
            v2f bm = *(const v2f*)(bp + k + koff);    // ds_load_b64 from LDS
            acc = __builtin_amdgcn_wmma_f32_16x16x4_f32(
                false, a, false, bm, (short)0, acc, false, false);
        }
        float* op = xw + ((size_t)b * NN + (size_t)mt * 16) * CH + (n0 + col);
#pragma unroll
        for (int r = 0; r < 8; ++r) {
            const int row = r + ((lane >> 4) << 3);
            op[(size_t)row * CH] = acc[r];
        }
    }
}

// ---------------------------------------------------------------------------
// GCN graph structure (constant-folded from the reference: sim_n <= 0 < 0.05
// always, so every value becomes -100 and top_k indices are [0,1,2,3]).
// ---------------------------------------------------------------------------
__device__ __forceinline__ void edge_rc(int e, int& row, int& col)
{
    const int i = (e < T1 * KTOP) ? e : e - T1 * KTOP;
    const int t = i / KTOP, j = i % KTOP;
    const int src = t * HW;
    const int dst = (t + 1) * HW + j;
    row = (e < T1 * KTOP) ? src : dst;
    col = (e < T1 * KTOP) ? dst : src;
}

__global__ void deg_init_kernel(float* __restrict__ deg)
{
    const int i = blockIdx.x * blockDim.x + threadIdx.x;
    if (i < BATCH * NN) deg[i] = 1.0f;                // self loop
}

__global__ void deg_edge_kernel(float* __restrict__ deg)
{
    const int b = blockIdx.x;
    const int e = threadIdx.x;
    if (e >= NEDGE) return;
    int row, col; edge_rc(e, row, col);
    atomicAdd(&deg[(size_t)b * NN + col], 1.0f);
}

__global__ void gcn_self_kernel(
    const float* __restrict__ xw, const float* __restrict__ deg,
    const float* __restrict__ gb, float* __restrict__ out)
{
    const size_t i = (size_t)blockIdx.x * blockDim.x + threadIdx.x;
    if (i >= (size_t)BATCH * NN * CH) return;
    const int c    = (int)(i % CH);
    const size_t n = i / CH;
    out[i] = xw[i] / deg[n] + gb[c];
}

__global__ void gcn_edge_kernel(
    const float* __restrict__ xw, const float* __restrict__ deg,
    float* __restrict__ out)
{
    const int e = blockIdx.x % NEDGE;
    const int b = blockIdx.x / NEDGE;
    const int c = threadIdx.x;
    int row, col; edge_rc(e, row, col);
    const float nrm = rsqrtf(deg[(size_t)b * NN + row]) *
                      rsqrtf(deg[(size_t)b * NN + col]);
    atomicAdd(&out[((size_t)b * NN + col) * CH + c],
              nrm * xw[((size_t)b * NN + row) * CH + c]);
}

// ---------------------------------------------------------------------------
// K5: temporal conv3d(3x1x1, pad 1) + BN on g[b,v,hw,c] -> out[(b*V+v), c, hw]
// Same structure as K1; A rows are channel-contiguous so A uses v2f loads.
// ---------------------------------------------------------------------------
__global__ __launch_bounds__(256) void conv_bn_up_kernel(
    const float* __restrict__ g, const float* __restrict__ w,
    const float* __restrict__ gamma, const float* __restrict__ beta,
    const float* __restrict__ mean, const float* __restrict__ var,
    float* __restrict__ out)
{
    __shared__ float lw[CH * CH * 3];
    if (threadIdx.x < 32) {
        tdm_load_1d((unsigned)(size_t)lw, w, CH * CH * 3);
        __builtin_amdgcn_s_wait_tensorcnt(0);
    }
    __syncthreads();

    const int lane = threadIdx.x & 31;
    const int wave = threadIdx.x >> 5;
    int blk = blockIdx.x;
    const int mg = blk % 7;  blk /= 7;
    const int v  = blk % VLEN; blk /= VLEN;
    const int b  = blk;
    const int m    = lane & 15;
    const int koff = (lane >> 4) << 1;
    const int col  = lane & 15;

    for (int tile = wave; tile < 7 * 4; tile += 8) {
        const int mt = mg * 7 + (tile >> 2);
        const int n0 = (tile & 3) * 16;
        const int p0 = mt * 16;
        v8f acc = {};
        for (int dt = 0; dt < 3; ++dt) {
            const int vv = v + dt - 1;
            if (vv < 0 || vv >= VLEN) continue;
            const float* ap = g + (((size_t)(b * VLEN + vv)) * HW + p0 + m) * CH;
            const float* wp = lw + (n0 + col) * CH * 3 + dt;
#pragma unroll
            for (int k = 0; k < CH; k += 4) {
                v2f a = *(const v2f*)(ap + k + koff); // contiguous channels
                v2f bm;
                bm.x = wp[(k + koff) * 3];
                bm.y = wp[(k + koff + 1) * 3];
                acc = __builtin_amdgcn_wmma_f32_16x16x4_f32(
                    false, a, false, bm, (short)0, acc, false, false);
            }
        }
        const int c = n0 + col;
        const float s = gamma[c] * rsqrtf(var[c] + BN_EPS);
        const float t = beta[c] - mean[c] * s;
        float* op = out + ((size_t)(b * VLEN + v) * CH + c) * HW + p0;
#pragma unroll
        for (int r = 0; r < 8; ++r) {
            const int row = r + ((lane >> 4) << 3);
            op[row] = acc[r] * s + t;
        }
    }
}

// ---------------------------------------------------------------------------
extern "C" void kernel_launch(void* const* d_in, const int* in_sizes, int n_in,
                              void* d_out, int out_size, void* d_ws, size_t ws_size,
                              hipStream_t stream)
{
    (void)in_sizes; (void)n_in; (void)out_size; (void)ws_size;
    const float* x  = (const float*)d_in[0];
    // d_in[1] = batch (int scalar) == 8, baked into launch geometry
    const float* dw = (const float*)d_in[2];
    const float* dg = (const float*)d_in[3];
    const float* db = (const float*)d_in[4];
    const float* dm = (const float*)d_in[5];
    const float* dv = (const float*)d_in[6];
    const float* gw = (const float*)d_in[7];
    const float* gb = (const float*)d_in[8];
    const float* uw = (const float*)d_in[9];
    const float* ug = (const float*)d_in[10];
    const float* ub = (const float*)d_in[11];
    const float* um = (const float*)d_in[12];
    const float* uv = (const float*)d_in[13];
    float* out = (float*)d_out;

    float* xd   = (float*)d_ws;                       // [B,NN,CH] 25.7 MB
    float* xw   = xd + (size_t)BATCH * NN * CH;       // [B,NN,CH] 25.7 MB
    float* deg  = xw + (size_t)BATCH * NN * CH;       // [B,NN]     0.4 MB
    float* gout = xd;                                 // reuse xd after K2

    conv_bn_down_kernel<<<dim3(BATCH * VLEN * 7), 256, 0, stream>>>(
        x, dw, dg, db, dm, dv, xd);
    gemm_xw_kernel<<<dim3(BATCH * (NN / 16 / 4)), 256, 0, stream>>>(xd, gw, xw);
    deg_init_kernel<<<dim3((BATCH * NN + 255) / 256), 256, 0, stream>>>(deg);
    deg_edge_kernel<<<dim3(BATCH), 128, 0, stream>>>(deg);
    gcn_self_kernel<<<dim3((BATCH * NN * CH) / 256), 256, 0, stream>>>(
        xw, deg, gb, gout);
    gcn_edge_kernel<<<dim3(BATCH * NEDGE), CH, 0, stream>>>(xw, deg, gout);
    conv_bn_up_kernel<<<dim3(BATCH * VLEN * 7), 256, 0, stream>>>(
        gout, uw, ug, ub, um, uv, out);
}